// GateRNN_66460323938980
// MI455X (gfx1250) — compile-verified
//
#include <hip/hip_runtime.h>
#include <math.h>

typedef __attribute__((ext_vector_type(2))) float v2f;
typedef __attribute__((ext_vector_type(8))) float v8f;

#define B_ 32
#define T_ 128
#define IN_ 64
#define D_ 128

// ---- Kernel 0: WhT[k][j] = Wh[j][k] (so WMMA B-fragments read contiguous rows)
__global__ void k_transpose(const float* __restrict__ Wh, float* __restrict__ WhT) {
  int idx = blockIdx.x * blockDim.x + threadIdx.x;
  if (idx < D_ * D_) {
    int r = idx >> 7, c = idx & (D_ - 1);
    WhT[c * D_ + r] = Wh[idx];
  }
}

// ---- Kernel 1: u = x @ Wx, x:[4096,64] Wx:[64,128] -> u:[4096,128], f32 WMMA
// one 16x16 output tile per wave; K=64 -> 16 v_wmma_f32_16x16x4_f32
__global__ void __launch_bounds__(256) k_xwx(const float* __restrict__ x,
                                             const float* __restrict__ Wx,
                                             float* __restrict__ u) {
  int w = threadIdx.x >> 5;
  int lane = threadIdx.x & 31;
  int g = blockIdx.x * 8 + w;          // 2048 tiles total
  int mtile = g >> 3, ntile = g & 7;
  int m0 = mtile * 16, n0 = ntile * 16;
  int lh = lane & 15;
  int khalf = (lane >= 16) ? 2 : 0;    // hi half-wave supplies K+2,K+3
  int mhalf = (lane >= 16) ? 8 : 0;    // C/D rows M and M+8 per VGPR

  v8f c = {};
  #pragma unroll
  for (int kc = 0; kc < IN_ / 4; ++kc) {
    int ka = kc * 4 + khalf;
    v2f a, b;
    a.x = x[(m0 + lh) * IN_ + ka];       // A[M=lh][K=ka]
    a.y = x[(m0 + lh) * IN_ + ka + 1];   // A[M=lh][K=ka+1]
    b.x = Wx[ka * D_ + n0 + lh];         // B[K=ka][N=lh]
    b.y = Wx[(ka + 1) * D_ + n0 + lh];
    c = __builtin_amdgcn_wmma_f32_16x16x4_f32(false, a, false, b, (short)0, c,
                                              false, false);
  }
  #pragma unroll
  for (int r = 0; r < 8; ++r)
    u[(m0 + r + mhalf) * D_ + n0 + lh] = c[r];
}

// ---- Kernel 2: sequential scan, one block per batch (32 independent chains).
// Stores y, h_final, and per-step d=gelu'(a), s=(h-1)g(1-g), g; then the
// re-applied final step's s2/d2 (reference re-applies step at h_final, x_last).
__global__ void __launch_bounds__(128) k_scan(
    const float* __restrict__ u, const float* __restrict__ Wh,
    const float* __restrict__ Wgate, float* __restrict__ hfinal,
    float* __restrict__ y, float* __restrict__ dbuf, float* __restrict__ sbuf,
    float* __restrict__ gbuf, float* __restrict__ s2, float* __restrict__ d2) {
  __shared__ float hs[D_];
  __shared__ float ns[D_];
  int b = blockIdx.x;
  int i = threadIdx.x;
  hs[i] = 0.0f;
  __syncthreads();
  for (int t = 0; t < T_; ++t) {
    float a = u[(b * T_ + t) * D_ + i];
    #pragma unroll 8
    for (int k = 0; k < D_; ++k) a = fmaf(hs[k], Wh[k * D_ + i], a);
    float phi = 0.5f * (1.0f + erff(a * 0.70710678118654752f)); // Phi(a)
    float pdf = 0.3989422804014327f * __expf(-0.5f * a * a);    // phi(a)
    ns[i] = a * phi;                                            // exact GELU
    dbuf[(t * B_ + b) * D_ + i] = phi + a * pdf;                // gelu'(a)
    __syncthreads();
    float z = 0.0f;
    #pragma unroll 8
    for (int k = 0; k < D_; ++k) z = fmaf(ns[k], Wgate[k * D_ + i], z);
    float gate = 1.0f / (1.0f + __expf(-z));
    float hp = hs[i];
    float hn = gate * hp + 1.0f - gate;
    y[(b * T_ + t) * D_ + i] = hn;
    gbuf[(t * B_ + b) * D_ + i] = gate;
    sbuf[(t * B_ + b) * D_ + i] = (hp - 1.0f) * gate * (1.0f - gate);
    __syncthreads();
    hs[i] = hn;
    __syncthreads();
  }
  hfinal[b * D_ + i] = hs[i];
  // extra application at (x_last, h_final) for dh_dWh factors
  float a = u[(b * T_ + (T_ - 1)) * D_ + i];
  #pragma unroll 8
  for (int k = 0; k < D_; ++k) a = fmaf(hs[k], Wh[k * D_ + i], a);
  float phi = 0.5f * (1.0f + erff(a * 0.70710678118654752f));
  float pdf = 0.3989422804014327f * __expf(-0.5f * a * a);
  ns[i] = a * phi;
  d2[b * D_ + i] = phi + a * pdf;
  __syncthreads();
  float z = 0.0f;
  #pragma unroll 8
  for (int k = 0; k < D_; ++k) z = fmaf(ns[k], Wgate[k * D_ + i], z);
  float gate = 1.0f / (1.0f + __expf(-z));
  s2[b * D_ + i] = (hs[i] - 1.0f) * gate * (1.0f - gate);
}

// ---- Kernel 3: Jacobians. One block per (t,b). P = (d-scaled Wgate^T) @ Wh^T,
// J[i,j] = delta_ij*g_i + s_i * P[i,j]. 8 waves x 8 j-tiles x 32 K-chunks WMMA.
__global__ void __launch_bounds__(256) k_jac(
    const float* __restrict__ Wgate, const float* __restrict__ WhT,
    const float* __restrict__ dbuf, const float* __restrict__ sbuf,
    const float* __restrict__ gbuf, float* __restrict__ Js) {
  int tb = blockIdx.x;                 // t*32 + b, matches Js [T,B,...] layout
  int w = threadIdx.x >> 5;            // itile 0..7
  int lane = threadIdx.x & 31;
  int lh = lane & 15;
  int khalf = (lane >= 16) ? 2 : 0;
  int ihalf = (lane >= 16) ? 8 : 0;
  int i0 = w * 16;
  const float* dv = dbuf + tb * D_;
  const float* sv = sbuf + tb * D_;
  const float* gv = gbuf + tb * D_;

  // Cache A fragments in VGPRs for this itile: A[i,k] = Wgate[k][i] * d[k]
  v2f Afr[32];
  #pragma unroll
  for (int kc = 0; kc < 32; ++kc) {
    int ka = kc * 4 + khalf;
    Afr[kc].x = Wgate[ka * D_ + i0 + lh] * dv[ka];
    Afr[kc].y = Wgate[(ka + 1) * D_ + i0 + lh] * dv[ka + 1];
  }
  float sreg[8], greg[8];
  #pragma unroll
  for (int r = 0; r < 8; ++r) {
    sreg[r] = sv[i0 + r + ihalf];
    greg[r] = gv[i0 + r + ihalf];
  }
  float* base = Js + (size_t)tb * (D_ * D_);
  for (int jt = 0; jt < 8; ++jt) {
    int j0 = jt * 16;
    v8f c = {};
    #pragma unroll
    for (int kc = 0; kc < 32; ++kc) {
      int ka = kc * 4 + khalf;
      v2f bf;
      bf.x = WhT[ka * D_ + j0 + lh];        // B[k][j] = Wh[j][k]
      bf.y = WhT[(ka + 1) * D_ + j0 + lh];
      c = __builtin_amdgcn_wmma_f32_16x16x4_f32(false, Afr[kc], false, bf,
                                                (short)0, c, false, false);
    }
    #pragma unroll
    for (int r = 0; r < 8; ++r) {
      int i = i0 + r + ihalf;
      int j = j0 + lh;
      float v = sreg[r] * c[r];
      if (i == j) v += greg[r];
      base[i * D_ + j] = v;
    }
  }
}

// ---- Kernel 4: dh_dWh[b,i,p,q] = s2[b,i] * hf[b,p] * d2[b,q] * Wgate[q,i]
// one block per (b,i); q = thread; loop p. Pure streaming stores.
__global__ void __launch_bounds__(128) k_dwh(
    const float* __restrict__ Wgate, const float* __restrict__ s2,
    const float* __restrict__ d2, const float* __restrict__ hfinal,
    float* __restrict__ out) {
  int b = blockIdx.x >> 7;
  int i = blockIdx.x & 127;
  int q = threadIdx.x;
  float eq = d2[b * D_ + q] * Wgate[q * D_ + i];
  float si = s2[b * D_ + i];
  const float* hf = hfinal + b * D_;
  float* row = out + ((size_t)(b * D_ + i)) * D_ * D_ + q;
  #pragma unroll 4
  for (int p = 0; p < D_; ++p)
    row[(size_t)p * D_] = si * hf[p] * eq;
}

extern "C" void kernel_launch(void* const* d_in, const int* in_sizes, int n_in,
                              void* d_out, int out_size, void* d_ws,
                              size_t ws_size, hipStream_t stream) {
  const float* x     = (const float*)d_in[0];  // [32,128,64]
  const float* Wx    = (const float*)d_in[1];  // [64,128]
  const float* Wgate = (const float*)d_in[2];  // [128,128]
  const float* Wh    = (const float*)d_in[3];  // [128,128]
  float* out = (float*)d_out;
  float* ws  = (float*)d_ws;

  // workspace layout (floats): total 2,121,728 (~8.5 MB)
  float* u    = ws;               // 524288  x@Wx for all (b,t)
  float* WhT  = u + 524288;       // 16384
  float* dbuf = WhT + 16384;      // 524288  gelu'(a) [t,b,i]
  float* sbuf = dbuf + 524288;    // 524288  (h-1)g(1-g)
  float* gbuf = sbuf + 524288;    // 524288  gate
  float* s2   = gbuf + 524288;    // 4096
  float* d2   = s2 + 4096;        // 4096

  // output layout: h_final | y | Js | dh_dWh
  float* hfinal = out;                       // 4096
  float* y      = out + 4096;                // 524288
  float* Js     = out + 4096 + 524288;       // 67108864
  float* dWh    = Js + (size_t)67108864;     // 67108864

  k_transpose<<<64, 256, 0, stream>>>(Wh, WhT);
  k_xwx<<<256, 256, 0, stream>>>(x, Wx, u);
  k_scan<<<32, 128, 0, stream>>>(u, Wh, Wgate, hfinal, y, dbuf, sbuf, gbuf, s2, d2);
  k_jac<<<4096, 256, 0, stream>>>(Wgate, WhT, dbuf, sbuf, gbuf, Js);
  k_dwh<<<4096, 128, 0, stream>>>(Wgate, s2, d2, hfinal, dWh);
}